// TruncatedConnection_58780922413164
// MI455X (gfx1250) — compile-verified
//
#include <hip/hip_runtime.h>
#include <stdint.h>

// Problem constants (fixed by the reference)
#define N_DATA   40320
#define N_TRUNC  10000
#define N_EDGE   322560
#define NB       4        // B * ENS batches after taking T=-1 slice
#define TDIM     2
#define FDIM     128
#define MAXTL    (N_EDGE / 16 + N_TRUNC)   // upper bound on 16-edge tiles (down)

typedef unsigned v4u __attribute__((ext_vector_type(4)));
typedef unsigned v8u __attribute__((ext_vector_type(8)));

// ---------------------------------------------------------------------------
// CDNA5 helpers
// ---------------------------------------------------------------------------
__device__ __forceinline__ unsigned lds_u32(const void* p) {
  // Generic LDS pointer: low 32 bits are the LDS byte address (aperture layout).
  return (unsigned)(uintptr_t)p;
}

// async global->LDS, GVS form: addr = SADDR(64) + VADDR(32); 16B per lane
__device__ __forceinline__ void async_load_b128(unsigned lds_byte_addr,
                                                unsigned voff_bytes,
                                                const void* sbase) {
  asm volatile("global_load_async_to_lds_b128 %0, %1, %2"
               :: "v"(lds_byte_addr), "v"(voff_bytes), "s"(sbase)
               : "memory");
}

template <int N>
__device__ __forceinline__ void wait_asynccnt() {
  asm volatile("s_wait_asynccnt %0" :: "i"(N) : "memory");
}

// TDM gather-mode load: fetch 16 rows of 128 f32 (selected by 16-bit indices)
// from a 2-D tensor at `gaddr` (row stride 128 elems) into LDS at lds_addr.
__device__ __forceinline__ void tdm_gather16(unsigned lds_addr, uint64_t gaddr,
                                             const v4u* __restrict__ idxp) {
  v4u g0;
  g0.x = 0x80000001u;                      // gather_mode=1, 16-bit idx, count=1
  g0.y = lds_addr;                         // LDS byte address of tile start
  g0.z = (unsigned)gaddr;                  // global_addr[31:0]
  g0.w = (unsigned)(gaddr >> 32) | 0x80000000u;  // global_addr[56:32] | type=2
  v8u g1;
  g1[0] = 0x00020000u;                     // data_size=2 (4B), no mask/pad/iter
  g1[1] = (unsigned)FDIM << 16;            // tensor_dim0 = 128
  g1[2] = (unsigned)N_DATA << 16;          // tensor_dim1[15:0]=40320 (dim0 hi=0)
  g1[3] = (unsigned)FDIM << 16;            // tile_dim0 = 128 (dim1 hi = 0)
  g1[4] = 16u;                             // tile_dim1 = 16 valid indices
  g1[5] = (unsigned)FDIM;                  // tensor_dim0_stride = 128 elems
  g1[6] = 0u;
  g1[7] = 0u;
  const v4u g2 = idxp[0];                  // row indices 0..7  (2 per dword)
  const v4u g3 = idxp[1];                  // row indices 8..15
  asm volatile("tensor_load_to_lds %0, %1, %2, %3"
               :: "s"(g0), "s"(g1), "s"(g2), "s"(g3)
               : "memory");
}

// ---------------------------------------------------------------------------
// 1) zero the four count/cursor arrays in one launch
// ---------------------------------------------------------------------------
__global__ void tc_zero4(int* __restrict__ a, int na, int* __restrict__ b, int nb_,
                         int* __restrict__ c, int nc, int* __restrict__ d, int nd) {
  const int i = blockIdx.x * blockDim.x + threadIdx.x;
  const int stride = gridDim.x * blockDim.x;
  for (int k = i; k < na;  k += stride) a[k] = 0;
  for (int k = i; k < nb_; k += stride) b[k] = 0;
  for (int k = i; k < nc;  k += stride) c[k] = 0;
  for (int k = i; k < nd;  k += stride) d[k] = 0;
}

// ---------------------------------------------------------------------------
// 2) histogram of destinations (both projections in one pass)
// ---------------------------------------------------------------------------
__global__ void tc_hist(const int* __restrict__ dn_dst, const int* __restrict__ up_dst,
                        int* __restrict__ cnt_dn, int* __restrict__ cnt_up, int ne) {
  int e = blockIdx.x * blockDim.x + threadIdx.x;
  if (e < ne) {
    atomicAdd(&cnt_dn[dn_dst[e]], 1);
    atomicAdd(&cnt_up[up_dst[e]], 1);
  }
}

// ---------------------------------------------------------------------------
// 3) two-level exclusive scan: block-local pass, 1-block sums pass, add pass.
//    Result: out[0..n-1] exclusive prefix, out[n] = total.
// ---------------------------------------------------------------------------
__global__ void tc_scan_block(const int* __restrict__ in, int* __restrict__ out,
                              int* __restrict__ sums, int n) {
  __shared__ int tmp[1024];
  const int tid = threadIdx.x;
  const int idx = blockIdx.x * 1024 + tid;
  const int v = (idx < n) ? in[idx] : 0;
  tmp[tid] = v;
  __syncthreads();
  for (int off = 1; off < 1024; off <<= 1) {
    const int t = (tid >= off) ? tmp[tid - off] : 0;
    __syncthreads();
    tmp[tid] += t;
    __syncthreads();
  }
  if (idx < n) out[idx] = tmp[tid] - v;        // block-local exclusive
  if (tid == 1023) sums[blockIdx.x] = tmp[1023];
}

__global__ void tc_scan_sums(int* __restrict__ sums, int nb_, int* __restrict__ tot) {
  __shared__ int tmp[1024];
  const int tid = threadIdx.x;
  const int v = (tid < nb_) ? sums[tid] : 0;
  tmp[tid] = v;
  __syncthreads();
  for (int off = 1; off < 1024; off <<= 1) {
    const int t = (tid >= off) ? tmp[tid - off] : 0;
    __syncthreads();
    tmp[tid] += t;
    __syncthreads();
  }
  if (tid < nb_) sums[tid] = tmp[tid] - v;     // exclusive block offsets
  if (tid == 1023) *tot = tmp[1023];           // grand total -> out[n]
}

__global__ void tc_scan_add(int* __restrict__ out, const int* __restrict__ sums, int n) {
  const int idx = blockIdx.x * 1024 + threadIdx.x;
  if (idx < n) out[idx] += sums[blockIdx.x];
}

// ---------------------------------------------------------------------------
// 4) scatter edge ids into CSR slots (order fixed up deterministically later)
// ---------------------------------------------------------------------------
__global__ void tc_scatter(const int* __restrict__ dn_dst, const int* __restrict__ up_dst,
                           const int* __restrict__ off_dn, const int* __restrict__ off_up,
                           int* __restrict__ cur_dn, int* __restrict__ cur_up,
                           int* __restrict__ sorted_dn, int* __restrict__ sorted_up, int ne) {
  int e = blockIdx.x * blockDim.x + threadIdx.x;
  if (e < ne) {
    int d = dn_dst[e];
    sorted_dn[off_dn[d] + atomicAdd(&cur_dn[d], 1)] = e;
    d = up_dst[e];
    sorted_up[off_up[d] + atomicAdd(&cur_up[d], 1)] = e;
  }
}

// ---------------------------------------------------------------------------
// 5a) per-row 16-edge tile counts (down projection)
// ---------------------------------------------------------------------------
__global__ void tc_tilecnt(const int* __restrict__ off, int* __restrict__ cnt, int nrows) {
  const int r = blockIdx.x * blockDim.x + threadIdx.x;
  if (r < nrows) cnt[r] = (off[r + 1] - off[r] + 15) >> 4;
}

// ---------------------------------------------------------------------------
// 5b) down: sort row edges (ascending id => deterministic) + emit TDM tiles:
//     16 u16 src indices (pad=0) + 16 f32 weights (pad=0.0) per tile.
// ---------------------------------------------------------------------------
__global__ void tc_sortpack_tiles(const int* __restrict__ offsets, int* __restrict__ sorted,
                                  const int* __restrict__ edge_src, const float* __restrict__ edge_w,
                                  const int* __restrict__ tile_off,
                                  unsigned short* __restrict__ idx16, float* __restrict__ wtile,
                                  int nrows) {
  const int r = blockIdx.x * blockDim.x + threadIdx.x;
  if (r >= nrows) return;
  const int beg = offsets[r], end = offsets[r + 1];
  for (int i = beg + 1; i < end; ++i) {
    const int key = sorted[i];
    int j = i - 1;
    while (j >= beg && sorted[j] > key) { sorted[j + 1] = sorted[j]; --j; }
    sorted[j + 1] = key;
  }
  const int tb  = tile_off[r];
  const int ntl = (end - beg + 15) >> 4;
  for (int t = 0; t < ntl; ++t) {
    for (int j = 0; j < 16; ++j) {
      const int i = beg + t * 16 + j;
      const size_t o = (size_t)(tb + t) * 16 + j;
      if (i < end) {
        const int e = sorted[i];
        idx16[o] = (unsigned short)edge_src[e];
        wtile[o] = edge_w[e];
      } else {
        idx16[o] = 0;       // valid row, weight 0 => no contribution
        wtile[o] = 0.0f;
      }
    }
  }
}

// ---------------------------------------------------------------------------
// 5c) up: sort row edges + pack (src, w) per edge for the async-LDS gather
// ---------------------------------------------------------------------------
__global__ void tc_sortpack(const int* __restrict__ offsets, int* __restrict__ sorted,
                            const int* __restrict__ edge_src, const float* __restrict__ edge_w,
                            int* __restrict__ psrc, float* __restrict__ pw, int nrows) {
  const int r = blockIdx.x * blockDim.x + threadIdx.x;
  if (r >= nrows) return;
  const int beg = offsets[r], end = offsets[r + 1];
  for (int i = beg + 1; i < end; ++i) {
    const int key = sorted[i];
    int j = i - 1;
    while (j >= beg && sorted[j] > key) { sorted[j + 1] = sorted[j]; --j; }
    sorted[j + 1] = key;
  }
  for (int i = beg; i < end; ++i) {
    const int e = sorted[i];
    psrc[i] = edge_src[e];
    pw[i]   = edge_w[e];
  }
}

// ---------------------------------------------------------------------------
// 6) down projection via TDM gather-mode:
//    coarse[b,r,:] = (sum_e w_e * x_t[b,src_e,:]) / sum_e w_e
//    Block = 4 wave32 (wave = batch). Each wave double-buffers 8KB LDS tiles
//    filled by tensor_load_to_lds (16 gathered rows per op, TENSORcnt-tracked).
// ---------------------------------------------------------------------------
__global__ void __launch_bounds__(128)
tc_down_gather(const float* __restrict__ x,
               const int* __restrict__ tile_off,
               const unsigned short* __restrict__ idx16,
               const float* __restrict__ wtile,
               float* __restrict__ coarse) {
  __shared__ float lds[NB * 2 * 16 * FDIM];   // 4 waves x 2 bufs x 8KB = 64KB

  const int r    = blockIdx.x;
  const int wave = threadIdx.x >> 5;          // == batch index
  const int lane = threadIdx.x & 31;

  const int tb = tile_off[r], te = tile_off[r + 1];

  const unsigned ldsb0 = lds_u32(&lds[0]) + (unsigned)(wave * 2) * 8192u;
  const unsigned ldsb1 = ldsb0 + 8192u;
  const uint64_t ga = (uint64_t)(uintptr_t)x +
                      (uint64_t)(wave * TDIM + 1) * (uint64_t)(N_DATA * FDIM) * 4ull;

  float ax = 0.f, ay = 0.f, az = 0.f, aw = 0.f, s = 0.f;

  if (tb < te)
    tdm_gather16(ldsb0, ga, (const v4u*)(idx16 + (size_t)tb * 16));

  int buf = 0;
  for (int t = tb; t < te; ++t) {
    if (t + 1 < te) {
      tdm_gather16(buf ? ldsb0 : ldsb1, ga, (const v4u*)(idx16 + (size_t)(t + 1) * 16));
      __builtin_amdgcn_s_wait_tensorcnt(1);   // in-order: tile t is in LDS
    } else {
      __builtin_amdgcn_s_wait_tensorcnt(0);
    }
    const float* base = &lds[(wave * 2 + buf) * (16 * FDIM) + lane * 4];
    const float* wt   = wtile + (size_t)t * 16;
#pragma unroll
    for (int j = 0; j < 16; ++j) {
      const float wj = wt[j];
      const float* p = base + j * FDIM;
      ax += p[0] * wj; ay += p[1] * wj; az += p[2] * wj; aw += p[3] * wj;
      s += wj;
    }
    buf ^= 1;
  }

  const float inv = (s == 0.0f) ? 1.0f : (1.0f / s);
  const size_t ob = (size_t)wave * (N_TRUNC * FDIM) + (size_t)r * FDIM + (size_t)(lane * 4);
  coarse[ob + 0] = ax * inv;
  coarse[ob + 1] = ay * inv;
  coarse[ob + 2] = az * inv;
  coarse[ob + 3] = aw * inv;
}

// ---------------------------------------------------------------------------
// 7) up projection: out[b,r,:] = (sum_e w_e * coarse[b,src_e,:]) / sum_e w_e
//    Source rows (L2-resident) staged through LDS with double-buffered
//    global_load_async_to_lds_b128 (ASYNCcnt-tracked, in-order completion).
// ---------------------------------------------------------------------------
__global__ void __launch_bounds__(128)
tc_up_gather(const float* __restrict__ coarse,
             const int* __restrict__ offsets,
             const int* __restrict__ psrc, const float* __restrict__ pw,
             float* __restrict__ out) {
  __shared__ float lds[NB * 2 * FDIM];        // 4 waves x 2 bufs x 512B = 4KB

  const int r     = blockIdx.x;
  const int batch = threadIdx.x >> 5;
  const int lane  = threadIdx.x & 31;

  const unsigned batch_voff = (unsigned)batch * (unsigned)(N_TRUNC * FDIM) * 4u;
  const int beg = offsets[r], end = offsets[r + 1];

  const unsigned buf0 = lds_u32(&lds[0]) + (unsigned)(batch * 2) * 512u + (unsigned)lane * 16u;
  const unsigned buf1 = buf0 + 512u;

  float ax = 0.f, ay = 0.f, az = 0.f, aw = 0.f, s = 0.f;

  if (beg < end) {
    const unsigned voff = batch_voff + (unsigned)psrc[beg] * 512u + (unsigned)lane * 16u;
    async_load_b128(buf0, voff, coarse);
  }
  int buf = 0;
  for (int i = beg; i < end; ++i) {
    const float w = pw[i];
    if (i + 1 < end) {
      const unsigned voff = batch_voff + (unsigned)psrc[i + 1] * 512u + (unsigned)lane * 16u;
      async_load_b128(buf ? buf0 : buf1, voff, coarse);
      wait_asynccnt<1>();
    } else {
      wait_asynccnt<0>();
    }
    const float* p = &lds[(batch * 2 + buf) * FDIM + lane * 4];
    ax += p[0] * w; ay += p[1] * w; az += p[2] * w; aw += p[3] * w;
    s += w;
    buf ^= 1;
  }

  const float inv = (s == 0.0f) ? 1.0f : (1.0f / s);
  const size_t ob = (size_t)batch * (N_DATA * FDIM) + (size_t)r * FDIM + (size_t)(lane * 4);
  out[ob + 0] = ax * inv;
  out[ob + 1] = ay * inv;
  out[ob + 2] = az * inv;
  out[ob + 3] = aw * inv;
}

// ---------------------------------------------------------------------------
// launch
// ---------------------------------------------------------------------------
static void launch_exscan(const int* in, int* out, int n, int* sums, hipStream_t stream) {
  const int nb_ = (n + 1023) >> 10;           // <= 40 blocks
  tc_scan_block<<<nb_, 1024, 0, stream>>>(in, out, sums, n);
  tc_scan_sums<<<1, 1024, 0, stream>>>(sums, nb_, out + n);
  tc_scan_add<<<nb_, 1024, 0, stream>>>(out, sums, n);
}

extern "C" void kernel_launch(void* const* d_in, const int* in_sizes, int n_in,
                              void* d_out, int out_size, void* d_ws, size_t ws_size,
                              hipStream_t stream) {
  (void)n_in; (void)out_size; (void)ws_size;

  const float* x       = (const float*)d_in[0];
  const int*   dn_src  = (const int*)d_in[1];
  const int*   dn_dst  = (const int*)d_in[2];
  const float* dn_w    = (const float*)d_in[3];
  const int*   up_src  = (const int*)d_in[4];
  const int*   up_dst  = (const int*)d_in[5];
  const float* up_w    = (const float*)d_in[6];
  float*       out     = (float*)d_out;

  const int ne = in_sizes[1];  // == N_EDGE

  // workspace carve-up (all chunk sizes multiples of 4 ints => 16B aligned)
  int* ws = (int*)d_ws;
  size_t o = 0;
  int*            cnt_dn  = ws + o;                     o += N_TRUNC;
  int*            off_dn  = ws + o;                     o += N_TRUNC + 4;
  int*            cur_dn  = ws + o;                     o += N_TRUNC;
  int*            srt_dn  = ws + o;                     o += N_EDGE;
  int*            cnt_tl  = ws + o;                     o += N_TRUNC;
  int*            tile_off= ws + o;                     o += N_TRUNC + 4;
  unsigned short* idx16   = (unsigned short*)(ws + o);  o += (size_t)MAXTL * 8;   // 16 u16 / tile
  float*          wtile   = (float*)(ws + o);           o += (size_t)MAXTL * 16;  // 16 f32 / tile
  int*            cnt_up  = ws + o;                     o += N_DATA;
  int*            off_up  = ws + o;                     o += N_DATA + 4;
  int*            cur_up  = ws + o;                     o += N_DATA;
  int*            srt_up  = ws + o;                     o += N_EDGE;
  int*            psrc_up = ws + o;                     o += N_EDGE;
  float*          pw_up   = (float*)(ws + o);           o += N_EDGE;
  int*            sums    = ws + o;                     o += 64;   // scan block sums
  float*          coarse  = (float*)(ws + o);           o += (size_t)NB * N_TRUNC * FDIM;

  const int eb = (ne + 255) / 256;

  tc_zero4<<<128, 256, 0, stream>>>(cnt_dn, N_TRUNC, cnt_up, N_DATA,
                                    cur_dn, N_TRUNC, cur_up, N_DATA);

  tc_hist<<<eb, 256, 0, stream>>>(dn_dst, up_dst, cnt_dn, cnt_up, ne);

  launch_exscan(cnt_dn, off_dn, N_TRUNC, sums, stream);
  launch_exscan(cnt_up, off_up, N_DATA,  sums, stream);

  tc_scatter<<<eb, 256, 0, stream>>>(dn_dst, up_dst, off_dn, off_up,
                                     cur_dn, cur_up, srt_dn, srt_up, ne);

  tc_tilecnt<<<(N_TRUNC + 255) / 256, 256, 0, stream>>>(off_dn, cnt_tl, N_TRUNC);
  launch_exscan(cnt_tl, tile_off, N_TRUNC, sums, stream);

  tc_sortpack_tiles<<<(N_TRUNC + 255) / 256, 256, 0, stream>>>(
      off_dn, srt_dn, dn_src, dn_w, tile_off, idx16, wtile, N_TRUNC);
  tc_sortpack<<<(N_DATA + 255) / 256, 256, 0, stream>>>(
      off_up, srt_up, up_src, up_w, psrc_up, pw_up, N_DATA);

  tc_down_gather<<<N_TRUNC, 128, 0, stream>>>(x, tile_off, idx16, wtile, coarse);
  tc_up_gather<<<N_DATA, 128, 0, stream>>>(coarse, off_up, psrc_up, pw_up, out);
}